// DDSL_spec_60902636258029
// MI455X (gfx1250) — compile-verified
//
#include <hip/hip_runtime.h>
#include <hip/hip_bf16.h>
#include <math.h>

// ---------------------------------------------------------------------------
// DDSL forward (j=2 simplices in 2D, RES=128x128, nch=1) for MI455X / gfx1250.
// Compute-bound (transcendental) kernel; WMMA F32_16X16X4_F32 computes the
// batched 2-D dot products sig[elem_vert, freq] = P . omega.
// ---------------------------------------------------------------------------

typedef float v2f __attribute__((ext_vector_type(2)));
typedef float v8f __attribute__((ext_vector_type(8)));

#define N_VERT   8192
#define N_ELEM   12800
#define R0       128
#define R1H      65
#define NFREQ    (R0 * R1H)          // 8320
#define FT_COUNT (NFREQ / 16)        // 520 freq tiles of 16
#define ET_PER_CHUNK 16              // element tiles (of 16 elems) per wave
#define CHUNKS   (N_ELEM / (16 * ET_PER_CHUNK))  // 50
#define TOTAL_WAVES (FT_COUNT * CHUNKS)          // 26000
#define WAVES_PER_BLOCK 8
#define BLOCKS  (TOTAL_WAVES / WAVES_PER_BLOCK)  // 3250

// ---------------------------------------------------------------------------
// Kernel 1: zero output + scratch accumulator
// ---------------------------------------------------------------------------
__global__ void ddsl_init(float* __restrict__ out, float* __restrict__ sumCD) {
    int i = blockIdx.x * blockDim.x + threadIdx.x;
    if (i < NFREQ * 2) out[i] = 0.0f;
    if (i == 0) *sumCD = 0.0f;
}

// ---------------------------------------------------------------------------
// Kernel 2: per-element prep. Gather the 3 vertices, compute CD = |cross| * D
// (C = j! * content = 2 * triangle area = |cross product|), reduce sum(CD).
// elemData layout per element (8 floats): p0x p0y p1x p1y p2x p2y cd pad
// ---------------------------------------------------------------------------
__global__ void ddsl_prep(const float* __restrict__ V,
                          const int*   __restrict__ E,
                          const float* __restrict__ D,
                          float* __restrict__ elemData,
                          float* __restrict__ sumCD) {
    int e = blockIdx.x * blockDim.x + threadIdx.x;
    float cd = 0.0f;
    if (e < N_ELEM) {
        int i0 = E[e * 3 + 0];
        int i1 = E[e * 3 + 1];
        int i2 = E[e * 3 + 2];
        float p0x = V[i0 * 2 + 0], p0y = V[i0 * 2 + 1];
        float p1x = V[i1 * 2 + 0], p1y = V[i1 * 2 + 1];
        float p2x = V[i2 * 2 + 0], p2y = V[i2 * 2 + 1];
        float cross = (p1x - p0x) * (p2y - p0y) - (p1y - p0y) * (p2x - p0x);
        cd = fabsf(cross) * D[e];
        float* ed = elemData + (size_t)e * 8;
        ed[0] = p0x; ed[1] = p0y;
        ed[2] = p1x; ed[3] = p1y;
        ed[4] = p2x; ed[5] = p2y;
        ed[6] = cd;  ed[7] = 0.0f;
    }
    // block-local reduction into one atomic per wave to limit contention
    float s = cd;
    s += __shfl_xor(s, 1, 32);
    s += __shfl_xor(s, 2, 32);
    s += __shfl_xor(s, 4, 32);
    s += __shfl_xor(s, 8, 32);
    s += __shfl_xor(s, 16, 32);
    if ((threadIdx.x & 31) == 0) atomicAdd(sumCD, s);
}

// ---------------------------------------------------------------------------
// Kernel 3: main. Each wave owns (freq tile of 16) x (chunk of 256 elements).
// Per 16-element tile: 3x V_WMMA_F32_16X16X4_F32 produce sig tiles for the
// three vertices in identical (M,N) register layouts; per-lane scalar math
// does the transcendental part; one atomicAdd pair per freq per wave at end.
// ---------------------------------------------------------------------------
__global__ __launch_bounds__(256) void ddsl_main(const float* __restrict__ elemData,
                                                 float* __restrict__ outF) {
    const int lane   = threadIdx.x & 31;
    const int wave   = blockIdx.x * WAVES_PER_BLOCK + (threadIdx.x >> 5);
    const int ft     = wave % FT_COUNT;   // freq tile
    const int chunk  = wave / FT_COUNT;   // element chunk
    if (chunk >= CHUNKS) return;          // wave-uniform; never taken at exact launch

    const int   laneLo = lane & 15;
    const int   fidx   = ft * 16 + laneLo;            // this lane's frequency
    const int   ix     = fidx / R1H;
    const int   iy     = fidx - ix * R1H;
    const float TWO_PI = 6.283185307179586f;
    const float wx = TWO_PI * (float)((ix < 64) ? ix : ix - 128);  // fftfreq * 2pi
    const float wy = TWO_PI * (float)iy;                            // rfftfreq * 2pi

    // B matrix (4x16, KxN): lanes 0-15 hold K=0/1 in VGPR0/1; lanes 16-31 hold
    // the zero K=2/3 padding (actual K is 2).
    const bool lo = (lane < 16);
    v2f B;
    B.x = lo ? wx : 0.0f;
    B.y = lo ? wy : 0.0f;

    float accRe = 0.0f, accIm = 0.0f;
    const int elemBase0 = chunk * (ET_PER_CHUNK * 16);

    for (int et = 0; et < ET_PER_CHUNK; ++et) {
        const int elemBase = elemBase0 + et * 16;
        const float* ed = elemData + (size_t)(elemBase + laneLo) * 8;
        // prefetch next tile's element data (speculative; dropped if OOB)
        __builtin_prefetch(ed + 128, 0, 0);

        const float p0x = ed[0], p0y = ed[1];
        const float p1x = ed[2], p1y = ed[3];
        const float p2x = ed[4], p2y = ed[5];
        const float cd  = ed[6];

        // A matrices (16x4, MxK): lanes 0-15 carry M=lane rows, K0/K1 in the
        // two VGPRs; lanes 16-31 carry the zero K2/K3 padding.
        v2f A0, A1, A2;
        A0.x = lo ? p0x : 0.0f;  A0.y = lo ? p0y : 0.0f;
        A1.x = lo ? p1x : 0.0f;  A1.y = lo ? p1y : 0.0f;
        A2.x = lo ? p2x : 0.0f;  A2.y = lo ? p2y : 0.0f;

        v8f z = {};
        // sig_v[M=elem, N=freq] = P_v . omega  — three tiles, identical layout
        v8f S0 = __builtin_amdgcn_wmma_f32_16x16x4_f32(false, A0, false, B, (short)0, z, false, false);
        v8f S1 = __builtin_amdgcn_wmma_f32_16x16x4_f32(false, A1, false, B, (short)0, z, false, false);
        v8f S2 = __builtin_amdgcn_wmma_f32_16x16x4_f32(false, A2, false, B, (short)0, z, false, false);

        const int mhi = (lane >= 16) ? 8 : 0;   // D layout: VGPR r -> M = r (+8 hi half)
#pragma unroll
        for (int r = 0; r < 8; ++r) {
            const float s0 = S0[r], s1 = S1[r], s2 = S2[r];
            const float d01 = s0 - s1;
            const float d12 = s1 - s2;
            const float d20 = s2 - s0;
            // denom_0 = -d01*d20, denom_1 = -d12*d01, denom_2 = -d20*d12
            // => 1/denom_v = -rcp(d01*d12*d20) * d_opposite : one reciprocal
            const float rcp = __builtin_amdgcn_rcpf(d01 * d12 * d20);
            float c0, sn0, c1, sn1, c2, sn2;
            __sincosf(s0, &sn0, &c0);
            __sincosf(s1, &sn1, &c1);
            __sincosf(s2, &sn2, &c2);
            // e^{-i s} = (cos s, -sin s); tmp = -rcp * (d12 e0 + d20 e1 + d01 e2)
            const float tre = -rcp * (d12 * c0  + d20 * c1  + d01 * c2);
            const float tim =  rcp * (d12 * sn0 + d20 * sn1 + d01 * sn2);
            const float cdm = __shfl(cd, r + mhi, 32);  // CD of element row M
            accRe += tre * cdm;
            accIm += tim * cdm;
        }
    }

    // fold the two M-halves (lane L <-> L+16 share the same frequency)
    accRe += __shfl_xor(accRe, 16, 32);
    accIm += __shfl_xor(accIm, 16, 32);

    // F_final = i^2 * RES^j * F_raw = -16384 * F_raw ; DC handled separately
    const float SCALE = -16384.0f;
    if (lane < 16 && fidx != 0) {
        atomicAdd(&outF[fidx * 2 + 0], SCALE * accRe);
        atomicAdd(&outF[fidx * 2 + 1], SCALE * accIm);
    }
}

// ---------------------------------------------------------------------------
// Kernel 4: DC term. F[0,0] = -sum(CD)/2 per reference, then * i^2 * 16384
// => +8192 * sum(CD) in both components (reference broadcasts to (nch,2)).
// ---------------------------------------------------------------------------
__global__ void ddsl_dc(const float* __restrict__ sumCD, float* __restrict__ out) {
    const float v = 8192.0f * (*sumCD);
    out[0] = v;
    out[1] = v;
}

// ---------------------------------------------------------------------------
extern "C" void kernel_launch(void* const* d_in, const int* in_sizes, int n_in,
                              void* d_out, int out_size, void* d_ws, size_t ws_size,
                              hipStream_t stream) {
    const float* V = (const float*)d_in[0];   // (8192, 2) f32
    const int*   E = (const int*)d_in[1];     // (12800, 3) int
    const float* D = (const float*)d_in[2];   // (12800, 1) f32
    float* out = (float*)d_out;               // (128, 65, 1, 2) flat

    float* elemData = (float*)d_ws;                   // 12800 * 8 floats
    float* sumCD    = elemData + (size_t)N_ELEM * 8;  // 1 float

    ddsl_init<<<(NFREQ * 2 + 255) / 256, 256, 0, stream>>>(out, sumCD);
    ddsl_prep<<<(N_ELEM + 255) / 256, 256, 0, stream>>>(V, E, D, elemData, sumCD);
    ddsl_main<<<BLOCKS, 256, 0, stream>>>(elemData, out);
    ddsl_dc<<<1, 1, 0, stream>>>(sumCD, out);
}